// AveragedHausdorffLoss_52072183497482
// MI455X (gfx1250) — compile-verified
//
#include <hip/hip_runtime.h>

typedef __attribute__((ext_vector_type(2))) float v2f;
typedef __attribute__((ext_vector_type(8))) float v8f;

#define NPTS 8192
#define DDIM 256
#define INF_BITS 0x7F800000u

// ---------------------------------------------------------------------------
// Kernel 1: per-row squared norms. One wave (32 lanes) per row of [NPTS, 256].
// ---------------------------------------------------------------------------
__global__ void sqnorm_kernel(const float* __restrict__ x, float* __restrict__ sq) {
    const int lane = threadIdx.x & 31;
    const int wave = threadIdx.x >> 5;
    const int row = blockIdx.x * 8 + wave;            // 8 waves per 256-thread block
    const float* p = x + (size_t)row * DDIM;
    float s = 0.0f;
#pragma unroll
    for (int j = 0; j < DDIM / 32; ++j) {
        float v = p[lane + 32 * j];
        s += v * v;
    }
#pragma unroll
    for (int m = 16; m >= 1; m >>= 1) s += __shfl_xor(s, m, 32);
    if (lane == 0) sq[row] = s;
}

// ---------------------------------------------------------------------------
// Kernel 2: initialize row/col min buffers to +inf (bit pattern).
// ---------------------------------------------------------------------------
__global__ void initmin_kernel(unsigned* __restrict__ mins, int n) {
    int i = blockIdx.x * blockDim.x + threadIdx.x;
    if (i < n) mins[i] = INF_BITS;
}

// ---------------------------------------------------------------------------
// Kernel 3: fused GEMM (f32 WMMA 16x16x4) + distance + row/col min reduction.
// Block = 256 threads = 8 waves arranged 4 (rows) x 2 (cols).
// Each wave computes a 32x32 output block: 4 accumulators, 2x2 fragment reuse.
// Workgroup macro-tile: 128 rows x 64 cols.
// ---------------------------------------------------------------------------
__global__ __launch_bounds__(256) void hausdorff_main_kernel(
    const float* __restrict__ set1, const float* __restrict__ set2,
    const float* __restrict__ sq1, const float* __restrict__ sq2,
    unsigned* __restrict__ rowmin, unsigned* __restrict__ colmin) {

    const int lane = threadIdx.x & 31;
    const int wave = threadIdx.x >> 5;
    const int wr = wave >> 1;          // 0..3  (row waves)
    const int wc = wave & 1;           // 0..1  (col waves)

    const int n0 = blockIdx.y * 128 + wr * 32;   // first output row of this wave
    const int m0 = blockIdx.x * 64 + wc * 32;    // first output col of this wave

    const int l16 = lane & 15;
    const int hi = lane >> 4;          // 0 for lanes 0-15, 1 for lanes 16-31
    const int koff = 2 * hi;           // K half-split per ISA f32 A/B layout

    // Per-lane base pointers for WMMA fragments (identical pattern for A and B,
    // since B = set2^T and set2 is row-major per point).
    const float* a0p = set1 + (size_t)(n0 + l16) * DDIM + koff;
    const float* a1p = a0p + (size_t)16 * DDIM;
    const float* b0p = set2 + (size_t)(m0 + l16) * DDIM + koff;
    const float* b1p = b0p + (size_t)16 * DDIM;

    v8f acc00 = {}, acc01 = {}, acc10 = {}, acc11 = {};

#pragma unroll 8
    for (int k = 0; k < DDIM; k += 4) {
        v2f a0 = *(const v2f*)(a0p + k);
        v2f a1 = *(const v2f*)(a1p + k);
        v2f b0 = *(const v2f*)(b0p + k);
        v2f b1 = *(const v2f*)(b1p + k);
        // 8 args: (neg_a, A, neg_b, B, c_mod, C, reuse_a, reuse_b)
        acc00 = __builtin_amdgcn_wmma_f32_16x16x4_f32(false, a0, false, b0, (short)0, acc00, false, false);
        acc01 = __builtin_amdgcn_wmma_f32_16x16x4_f32(false, a0, false, b1, (short)0, acc01, false, false);
        acc10 = __builtin_amdgcn_wmma_f32_16x16x4_f32(false, a1, false, b0, (short)0, acc10, false, false);
        acc11 = __builtin_amdgcn_wmma_f32_16x16x4_f32(false, a1, false, b1, (short)0, acc11, false, false);
    }

    // Squared norms for epilogue. C/D layout: lane holds col N=l16,
    // VGPR r holds row M = r + 8*hi (per 16x16 tile).
    float sqb0 = sq2[m0 + l16];
    float sqb1 = sq2[m0 + 16 + l16];
    float sqa[2][8];
#pragma unroll
    for (int i = 0; i < 2; ++i)
#pragma unroll
        for (int r = 0; r < 8; ++r)
            sqa[i][r] = sq1[n0 + 16 * i + r + 8 * hi];

    // Distances + reductions.
    float colm0 = __builtin_inff();
    float colm1 = __builtin_inff();
#pragma unroll
    for (int i = 0; i < 2; ++i) {
        const v8f accL = (i == 0) ? acc00 : acc10;
        const v8f accR = (i == 0) ? acc01 : acc11;
#pragma unroll
        for (int r = 0; r < 8; ++r) {
            float dL = __builtin_sqrtf(__builtin_fmaxf(sqa[i][r] + sqb0 - 2.0f * accL[r], 0.0f));
            float dR = __builtin_sqrtf(__builtin_fmaxf(sqa[i][r] + sqb1 - 2.0f * accR[r], 0.0f));
            colm0 = __builtin_fminf(colm0, dL);
            colm1 = __builtin_fminf(colm1, dR);
            // Row min: reduce across the 16 lanes of each half-wave
            // (lanes 0-15 = row n0+16i+r, lanes 16-31 = row n0+16i+r+8).
            float rm = __builtin_fminf(dL, dR);
#pragma unroll
            for (int m = 8; m >= 1; m >>= 1) rm = __builtin_fminf(rm, __shfl_xor(rm, m, 32));
            if (l16 == 0) {
                atomicMin(rowmin + (n0 + 16 * i + r + 8 * hi), __float_as_uint(rm));
            }
        }
    }
    // Col min: lanes L and L+16 hold the same column; fold halves, then lanes
    // 0-15 commit one atomic per column.
    colm0 = __builtin_fminf(colm0, __shfl_xor(colm0, 16, 32));
    colm1 = __builtin_fminf(colm1, __shfl_xor(colm1, 16, 32));
    if (hi == 0) {
        atomicMin(colmin + (m0 + l16), __float_as_uint(colm0));
        atomicMin(colmin + (m0 + 16 + l16), __float_as_uint(colm1));
    }
}

// ---------------------------------------------------------------------------
// Kernel 4: reduce the two min vectors to mean(rowmin) + mean(colmin).
// ---------------------------------------------------------------------------
__global__ void finalize_kernel(const unsigned* __restrict__ rowmin,
                                const unsigned* __restrict__ colmin,
                                float* __restrict__ out) {
    __shared__ float buf[256];
    const int t = threadIdx.x;

    float s1 = 0.0f;
    for (int j = 0; j < NPTS / 256; ++j) s1 += __uint_as_float(rowmin[t + 256 * j]);
    float s2 = 0.0f;
    for (int j = 0; j < NPTS / 256; ++j) s2 += __uint_as_float(colmin[t + 256 * j]);

    buf[t] = s1 + s2;
    __syncthreads();
    for (int stride = 128; stride >= 1; stride >>= 1) {
        if (t < stride) buf[t] += buf[t + stride];
        __syncthreads();
    }
    if (t == 0) out[0] = buf[0] / (float)NPTS;
}

// ---------------------------------------------------------------------------
// Harness entry point.
// ---------------------------------------------------------------------------
extern "C" void kernel_launch(void* const* d_in, const int* in_sizes, int n_in,
                              void* d_out, int out_size, void* d_ws, size_t ws_size,
                              hipStream_t stream) {
    const float* set1 = (const float*)d_in[0];
    const float* set2 = (const float*)d_in[1];
    float* out = (float*)d_out;

    // Workspace layout: sq1[8192] f32 | sq2[8192] f32 | rowmin[8192] u32 | colmin[8192] u32
    float* sq1 = (float*)d_ws;
    float* sq2 = sq1 + NPTS;
    unsigned* rowmin = (unsigned*)(sq2 + NPTS);
    unsigned* colmin = rowmin + NPTS;

    sqnorm_kernel<<<NPTS / 8, 256, 0, stream>>>(set1, sq1);
    sqnorm_kernel<<<NPTS / 8, 256, 0, stream>>>(set2, sq2);
    initmin_kernel<<<(2 * NPTS + 255) / 256, 256, 0, stream>>>(rowmin, 2 * NPTS);

    dim3 grid(NPTS / 64, NPTS / 128);   // x: col macro-tiles, y: row macro-tiles
    hausdorff_main_kernel<<<grid, 256, 0, stream>>>(set1, set2, sq1, sq2, rowmin, colmin);

    finalize_kernel<<<1, 256, 0, stream>>>(rowmin, colmin, out);
}